// BOTNet_45981919871463
// MI455X (gfx1250) — compile-verified
//
#include <hip/hip_runtime.h>
#include <hip/hip_bf16.h>

// ---------------- problem constants (match reference) ----------------
#define NN     10000
#define EE     160000
#define NEL    10
#define CC     64
#define NB     8
#define SHD    9
#define MLPH   16
#define TT     2
#define R_MAX  5.0f
#define CSH    (CC*SHD)          // 576
#define INV_AVG 0.03125f         // 1/32

typedef float v2f __attribute__((ext_vector_type(2)));
typedef float v8f __attribute__((ext_vector_type(8)));
typedef unsigned int v4u __attribute__((ext_vector_type(4)));
typedef int v4i __attribute__((ext_vector_type(4)));
typedef int v8i __attribute__((ext_vector_type(8)));

#if defined(__has_builtin)
#  if __has_builtin(__builtin_amdgcn_tensor_load_to_lds)
#    define HAVE_TDM 1
#  else
#    define HAVE_TDM 0
#  endif
#else
#  define HAVE_TDM 0
#endif

__device__ __forceinline__ float silu(float x) {
    return x / (1.0f + __expf(-x));
}

// ---------------------------------------------------------------------
// Kernel 1: per-edge geometry -> radial features (8) + spherical harmonics (9)
// E = 160000 = 625 * 256 exactly.
// ---------------------------------------------------------------------
__global__ void edge_geom_kernel(const float* __restrict__ pos,
                                 const float* __restrict__ shifts,
                                 const int*   __restrict__ ei,
                                 float* __restrict__ ef,
                                 float* __restrict__ shv)
{
    const int e = blockIdx.x * 256 + threadIdx.x;
    const int snd = ei[e];
    const int rcv = ei[EE + e];

    const float vx = pos[rcv*3+0] - pos[snd*3+0] + shifts[e*3+0];
    const float vy = pos[rcv*3+1] - pos[snd*3+1] + shifts[e*3+1];
    const float vz = pos[rcv*3+2] - pos[snd*3+2] + shifts[e*3+2];
    const float r   = sqrtf(vx*vx + vy*vy + vz*vz) + 1e-9f;
    const float inv = 1.0f / r;
    const float x = vx*inv, y = vy*inv, z = vz*inv;

    const float s3 = 1.73205080757f, s5 = 2.2360679775f, s15 = 3.87298334621f;
    shv[e*SHD+0] = 1.0f;
    shv[e*SHD+1] = s3*x;
    shv[e*SHD+2] = s3*y;
    shv[e*SHD+3] = s3*z;
    shv[e*SHD+4] = s15*x*y;
    shv[e*SHD+5] = s15*y*z;
    shv[e*SHD+6] = 0.5f*s5*(3.0f*z*z - 1.0f);
    shv[e*SHD+7] = s15*x*z;
    shv[e*SHD+8] = 0.5f*s15*(x*x - y*y);

    // polynomial cutoff, p = 6 : fc = 1 - 28 x^6 + 48 x^7 - 21 x^8
    float xr = r * (1.0f/R_MAX);
    xr = fminf(xr, 1.0f);
    const float x2 = xr*xr, x3 = x2*xr, x6 = x3*x3, x7 = x6*xr, x8 = x7*xr;
    float fc = 1.0f - 28.0f*x6 + 48.0f*x7 - 21.0f*x8;
    fc = (xr < 1.0f) ? fc : 0.0f;

    const float base = 0.632455532034f * inv * fc;   // sqrt(2/5)/r * fc
    const float pr = 3.14159265358979f * r * (1.0f/R_MAX);
    #pragma unroll
    for (int n = 1; n <= NB; ++n)
        ef[e*NB + (n-1)] = base * __sinf((float)n * pr);
}

// ---------------------------------------------------------------------
// Kernel 2a: h = node_attrs @ W_embed  (one thread per (n,c); N*C = 640000)
// ---------------------------------------------------------------------
__global__ void embed_h_kernel(const float* __restrict__ attrs,
                               const float* __restrict__ Wemb,
                               float* __restrict__ h)
{
    const int i = blockIdx.x * 256 + threadIdx.x;
    const int n = i >> 6, c = i & 63;
    float acc = 0.0f;
    #pragma unroll
    for (int k = 0; k < NEL; ++k)
        acc += attrs[n*NEL + k] * Wemb[k*CC + c];
    h[i] = acc;
}

// Kernel 2b: reps[0] = node_attrs @ atomic_energies
__global__ void embed_e0_kernel(const float* __restrict__ attrs,
                                const float* __restrict__ ae,
                                float* __restrict__ out)
{
    const int n = blockIdx.x * 256 + threadIdx.x;
    if (n >= NN) return;
    float acc = 0.0f;
    #pragma unroll
    for (int k = 0; k < NEL; ++k)
        acc += attrs[n*NEL + k] * ae[k];
    out[n] = acc;
}

// ---------------------------------------------------------------------
// Generic fp32 WMMA GEMM:  Out[M x 64] = A[M x K] @ B[K x 64] (+ Cin)
// grid = (M/16, 64/16), block = 32 (one wave per 16x16 tile).
// V_WMMA_F32_16X16X4_F32: exact fp32 math, matches reference.
// ---------------------------------------------------------------------
__global__ void gemm_wmma_f32(const float* __restrict__ A,
                              const float* __restrict__ B,
                              const float* __restrict__ Cin,
                              float* __restrict__ Out,
                              int K)
{
    const int lane = threadIdx.x;
    const int half = lane >> 4;      // 0: K offsets {0,1}; 1: {2,3}
    const int idx  = lane & 15;
    const int row_base = blockIdx.x * 16;
    const int col      = blockIdx.y * 16 + idx;

    v8f acc = {};
    const size_t arow = (size_t)(row_base + idx) * (size_t)K;
    for (int k0 = 0; k0 < K; k0 += 4) {
        const int ka = k0 + 2*half;
        v2f a, b;
        a.x = A[arow + ka];
        a.y = A[arow + ka + 1];
        b.x = B[(size_t)ka * CC + col];
        b.y = B[(size_t)(ka+1) * CC + col];
        acc = __builtin_amdgcn_wmma_f32_16x16x4_f32(false, a, false, b,
                                                    (short)0, acc, false, false);
    }
    #pragma unroll
    for (int v = 0; v < 8; ++v) {
        const int row = row_base + v + 8*half;
        float r = acc[v];
        if (Cin) r += Cin[(size_t)row * CC + col];
        Out[(size_t)row * CC + col] = r;
    }
}

// ---------------------------------------------------------------------
// Kernel 4 (fused hotspot): per-edge radial MLP (WMMA) -> message -> scatter.
// One wave owns 16 edges. E = 160000 = 10000 waves = 2500 blocks * 4 waves.
//   W_r2 (16x576 fp32 = 36.9 KB) staged into LDS once per block by the
//   Tensor Data Mover (tensor_load_to_lds + s_wait_tensorcnt), then:
//   hidden(16x16) = silu(ef(16x8) @ W_r1(8x16))           (VALU, into LDS)
//   tp(16x576)    = hidden @ W_r2_lds                     (36 tiles * 4 WMMA)
//   agg[rcv] += tp * h_up[snd][c] * sh[s] / 32            (f32 atomics, L2)
// msg (368 MB if materialized) never touches memory.
// ---------------------------------------------------------------------
#define WPB 4
__global__ void edge_msg_kernel(const float* __restrict__ ef,
                                const float* __restrict__ shv,
                                const int*   __restrict__ ei,
                                const float* __restrict__ hup,
                                const float* __restrict__ Wr1,
                                const float* __restrict__ Wr2,
                                float* __restrict__ agg)
{
    __shared__ float s_wr2[MLPH][CSH];       // 36864 B, TDM-filled
    __shared__ float s_hid[WPB][16][MLPH];   // hidden activations
    __shared__ float s_hup[WPB][16][CC];     // gathered h_up[snd]
    __shared__ float s_sh [WPB][16][SHD];    // spherical harmonics
    __shared__ int   s_rcv[WPB][16];

    const int lane = threadIdx.x & 31;
    const int wv   = threadIdx.x >> 5;
    const int half = lane >> 4;
    const int idx  = lane & 15;
    const int e0   = (blockIdx.x * WPB + wv) * 16;

    // ---- stage W_r2 into LDS ----
#if HAVE_TDM
    if (wv == 0) {
        const unsigned long long ga = (unsigned long long)(uintptr_t)Wr2;
        const unsigned int lds_addr = (unsigned int)(uintptr_t)(&s_wr2[0][0]);
        // D# group0: count=1 | lds_addr | global_addr[56:0] | type=2
        v4u g0;
        g0.x = 1u;                                   // count = 1 valid descriptor
        g0.y = lds_addr;                             // LDS byte address
        g0.z = (unsigned int)(ga & 0xFFFFFFFFu);     // global_addr[31:0]
        g0.w = (unsigned int)(((ga >> 32) & 0x01FFFFFFu) | (2u << 30)); // [56:32] | type=2
        // D# group1: 2-D tensor 576 x 16 fp32, tile = whole tensor
        v8i g1;
        g1[0] = (int)(2u << 16);                     // workgroup_mask=0, data_size=4B
        g1[1] = (int)((576u & 0xFFFFu) << 16);       // tensor_dim0[15:0] in bits[63:48]
        g1[2] = (int)((576u >> 16) | (16u << 16));   // tensor_dim0[31:16] | tensor_dim1[15:0]
        g1[3] = (int)(576u << 16);                   // tensor_dim1[31:16]=0 | tile_dim0=576
        g1[4] = 16;                                  // tile_dim1=16, tile_dim2=0
        g1[5] = 576;                                 // tensor_dim0_stride[31:0]
        g1[6] = 0;                                   // stride0[47:32]=0, stride1[15:0]=0
        g1[7] = 0;
        v4i z4 = {0, 0, 0, 0};
#if defined(__clang_major__) && (__clang_major__ >= 23)
        v8i z8 = {0, 0, 0, 0, 0, 0, 0, 0};
        __builtin_amdgcn_tensor_load_to_lds(g0, g1, z4, z4, z8, 0);
#else
        __builtin_amdgcn_tensor_load_to_lds(g0, g1, z4, z4, 0);
#endif
        __builtin_amdgcn_s_wait_tensorcnt(0);
    }
#else
    for (int i = threadIdx.x; i < MLPH * CSH; i += 32 * WPB)
        (&s_wr2[0][0])[i] = Wr2[i];
#endif

    // ---- load phase: lane (half,idx) services edge m = idx ----
    const int e   = e0 + idx;
    const int snd = ei[e];
    const int rcv = ei[EE + e];
    if (half == 0) s_rcv[wv][idx] = rcv;

    // gather h_up[snd]: each half loads 32 of 64 channels as 8 x float4
    {
        const float4* src = (const float4*)(hup + (size_t)snd * CC + half * 32);
        float4* dst = (float4*)(&s_hup[wv][idx][half * 32]);
        #pragma unroll
        for (int j = 0; j < 8; ++j) dst[j] = src[j];
    }

    if (half == 0) {
        #pragma unroll
        for (int s = 0; s < SHD; ++s)
            s_sh[wv][idx][s] = shv[(size_t)e*SHD + s];
    }

    // hidden = silu(ef @ W_r1): each lane computes 8 of the 16 hidden units
    float efr[NB];
    {
        const float4* src = (const float4*)(ef + (size_t)e * NB);
        float4 a0 = src[0], a1 = src[1];
        efr[0]=a0.x; efr[1]=a0.y; efr[2]=a0.z; efr[3]=a0.w;
        efr[4]=a1.x; efr[5]=a1.y; efr[6]=a1.z; efr[7]=a1.w;
    }
    #pragma unroll
    for (int hh = 0; hh < 8; ++hh) {
        const int hc = half*8 + hh;
        float acc = 0.0f;
        #pragma unroll
        for (int j = 0; j < NB; ++j)
            acc += efr[j] * Wr1[j*MLPH + hc];
        s_hid[wv][idx][hc] = silu(acc);
    }

    __syncthreads();

    // ---- GEMM + scatter phase: tp(16x576) in 36 column tiles ----
    for (int nt = 0; nt < CSH/16; ++nt) {
        v8f acc = {};
        const int col = nt*16 + idx;        // col = c*9 + s
        #pragma unroll
        for (int k0 = 0; k0 < MLPH; k0 += 4) {
            const int ka = k0 + 2*half;
            v2f a, b;
            a.x = s_hid[wv][idx][ka];
            a.y = s_hid[wv][idx][ka + 1];
            b.x = s_wr2[ka][col];
            b.y = s_wr2[ka + 1][col];
            acc = __builtin_amdgcn_wmma_f32_16x16x4_f32(false, a, false, b,
                                                        (short)0, acc, false, false);
        }
        const int c = col / SHD;
        const int s = col - SHD*c;
        #pragma unroll
        for (int v = 0; v < 8; ++v) {
            const int row = v + 8*half;     // edge within tile
            const float msg = acc[v] * s_hup[wv][row][c] * s_sh[wv][row][s] * INV_AVG;
            atomicAdd(&agg[(size_t)s_rcv[wv][row] * CSH + col], msg);
        }
    }
}

// ---------------------------------------------------------------------
// Readouts
// ---------------------------------------------------------------------
__global__ void read_lin_kernel(const float* __restrict__ h,
                                const float* __restrict__ w,
                                float* __restrict__ out)
{
    const int n = blockIdx.x * 256 + threadIdx.x;
    if (n >= NN) return;
    float acc = 0.0f;
    #pragma unroll 8
    for (int c = 0; c < CC; ++c) acc += h[(size_t)n*CC + c] * w[c];
    out[n] = acc;
}

__global__ void read_mlp_kernel(const float* __restrict__ h,
                                const float* __restrict__ W1,
                                const float* __restrict__ w2,
                                float* __restrict__ out)
{
    const int n = blockIdx.x * 256 + threadIdx.x;
    if (n >= NN) return;
    float acc = 0.0f;
    #pragma unroll
    for (int j = 0; j < MLPH; ++j) {
        float t = 0.0f;
        #pragma unroll 8
        for (int c = 0; c < CC; ++c)
            t += h[(size_t)n*CC + c] * W1[c*MLPH + j];
        acc += silu(t) * w2[j];
    }
    out[2*NN + n] = acc;
}

// ---------------------------------------------------------------------
extern "C" void kernel_launch(void* const* d_in, const int* in_sizes, int n_in,
                              void* d_out, int out_size, void* d_ws, size_t ws_size,
                              hipStream_t stream)
{
    const float* pos    = (const float*)d_in[0];
    const float* attrs  = (const float*)d_in[1];
    const float* shifts = (const float*)d_in[2];
    const int*   ei     = (const int*)  d_in[3];
    const float* ae     = (const float*)d_in[4];
    const float* Wemb   = (const float*)d_in[5];
    const float* Wup    = (const float*)d_in[6];
    const float* Wr1    = (const float*)d_in[7];
    const float* Wr2    = (const float*)d_in[8];
    const float* Wmix   = (const float*)d_in[9];
    const float* Wself  = (const float*)d_in[10];
    const float* wread  = (const float*)d_in[11];
    const float* Wmlp1  = (const float*)d_in[12];
    const float* wmlp2  = (const float*)d_in[13];
    float* out = (float*)d_out;

    // workspace layout (fp32)
    float* ws    = (float*)d_ws;
    float* ef    = ws;  ws += (size_t)EE * NB;    // 5.1 MB
    float* shv   = ws;  ws += (size_t)EE * SHD;   // 5.8 MB
    float* hA    = ws;  ws += (size_t)NN * CC;
    float* hB    = ws;  ws += (size_t)NN * CC;
    float* hup   = ws;  ws += (size_t)NN * CC;
    float* hself = ws;  ws += (size_t)NN * CC;
    float* agg   = ws;  ws += (size_t)NN * CSH;   // 23 MB -> L2 resident

    edge_geom_kernel<<<EE/256, 256, 0, stream>>>(pos, shifts, ei, ef, shv);
    embed_h_kernel<<<(NN*CC)/256, 256, 0, stream>>>(attrs, Wemb, hA);
    embed_e0_kernel<<<(NN+255)/256, 256, 0, stream>>>(attrs, ae, out);

    float* h  = hA;
    float* hn = hB;
    const dim3 gT(NN/16, CC/16);   // 625 x 4 tiles, wave32 per tile

    for (int t = 0; t < TT; ++t) {
        hipMemsetAsync(agg, 0, (size_t)NN * CSH * sizeof(float), stream);

        gemm_wmma_f32<<<gT, 32, 0, stream>>>(h, Wup   + (size_t)t*CC*CC,  nullptr, hup,   CC);
        gemm_wmma_f32<<<gT, 32, 0, stream>>>(h, Wself + (size_t)t*CC*CC,  nullptr, hself, CC);

        edge_msg_kernel<<<EE/(16*WPB), 32*WPB, 0, stream>>>(
            ef, shv, ei, hup,
            Wr1 + (size_t)t*NB*MLPH,
            Wr2 + (size_t)t*MLPH*CSH,
            agg);

        gemm_wmma_f32<<<gT, 32, 0, stream>>>(agg, Wmix + (size_t)t*CSH*CC, hself, hn, CSH);

        if (t == 0)
            read_lin_kernel<<<(NN+255)/256, 256, 0, stream>>>(hn, wread, out + NN);
        else
            read_mlp_kernel<<<(NN+255)/256, 256, 0, stream>>>(hn, Wmlp1, wmlp2, out);

        float* tmp = h; h = hn; hn = tmp;
    }
}